// ReLUTransformer_24026047053925
// MI455X (gfx1250) — compile-verified
//
#include <hip/hip_runtime.h>

typedef __attribute__((ext_vector_type(2))) float v2f;
typedef __attribute__((ext_vector_type(8))) float v8f;

// One wave handles 16 elements as a 16x16 diagonal block.
//   lower = diag(beta)  @ lb            -> WMMA outer product, take diagonal
//   upper = diag(lmbda) @ ub + diag(mu) -> WMMA outer product with C = diag(mu)
// V_WMMA_F32_16X16X4_F32 layouts (wave32):
//   A (16x4):  lanes 0-15 hold M=lane; VGPR0 = K0 (we use only K0)
//   B (4x16):  VGPR0 lanes 0-15 = row K0, N=lane (K>=1 zeroed)
//   C/D (16x16): lanes 0-15: (M=r, N=lane); lanes 16-31: (M=8+r, N=lane-16)
// Diagonal (i,i): lanes 0-7 at r=lane; lanes 24-31 at r=lane-24.
__global__ __launch_bounds__(256) void relu_bounds_wmma(
    const float* __restrict__ bounds, float* __restrict__ out, int n) {
  const int lane = threadIdx.x & 31;
  const int wave = threadIdx.x >> 5;
  const int base = (blockIdx.x * 8 + wave) * 16;

  const int e  = base + (lane & 15);
  const int ec = (e < n) ? e : (n - 1);   // clamp loads; keeps EXEC uniform for WMMA

  const float lb = bounds[ec];        // row 0 = lower
  const float ub = bounds[n + ec];    // row 1 = upper

  // ReLU relaxation masks (selects only -> no EXEC divergence before WMMA)
  const bool ind2 = (lb >= 0.0f);
  const bool ind3 = (ub > 0.0f) && (lb < 0.0f);
  const bool ind4 = ind3 && (ub > -lb);

  const float diff   = ind3 ? (ub - lb) : 1.0f;
  const float beta   = (ind2 || ind4) ? 1.0f : 0.0f;
  const float lmbda  = ind2 ? 1.0f : (ind3 ? (ub / diff) : 0.0f);
  const float mu     = ind3 ? ((-lb * ub) / diff) : 0.0f;
  const float new_lb = (ind2 || ind4) ? lb : 0.0f;
  const float new_ub = (ind2 || ind3) ? ub : 0.0f;

  const bool half0 = (lane < 16);
  v2f aL; aL.x = half0 ? beta  : 0.0f; aL.y = 0.0f;   // A column K0 = beta
  v2f bL; bL.x = half0 ? lb    : 0.0f; bL.y = 0.0f;   // B row    K0 = lb
  v2f aU; aU.x = half0 ? lmbda : 0.0f; aU.y = 0.0f;   // A column K0 = lmbda
  v2f bU; bU.x = half0 ? ub    : 0.0f; bU.y = 0.0f;   // B row    K0 = ub

  const bool owner = (lane < 8) || (lane >= 24);      // lanes holding the diagonal
  const int  r     = lane & 7;

  v8f cz = {};
  v8f cU;
#pragma unroll
  for (int k = 0; k < 8; ++k) cU[k] = (owner && (r == k)) ? mu : 0.0f;  // C = diag(mu)

  // 8 args: (neg_a, A, neg_b, B, c_mod, C, reuse_a, reuse_b)
  v8f dL = __builtin_amdgcn_wmma_f32_16x16x4_f32(false, aL, false, bL,
                                                 (short)0, cz, false, false);
  v8f dU = __builtin_amdgcn_wmma_f32_16x16x4_f32(false, aU, false, bU,
                                                 (short)0, cU, false, false);

  // Extract our diagonal entry (static unroll -> v_cndmask chain, no scratch)
  float lower = 0.0f, upper = 0.0f;
#pragma unroll
  for (int k = 0; k < 8; ++k) {
    if (r == k) { lower = dL[k]; upper = dU[k]; }
  }

  // Keep tighter of relaxed vs back-substituted bounds; owner lanes already
  // hold new_lb/new_ub for their element (they loaded base + (lane&15)).
  if (owner && (e < n)) {
    const float olb = (lower > new_lb) ? lower : new_lb;
    const float oub = (upper < new_ub) ? upper : new_ub;
    out[e]     = olb;   // out row 0
    out[n + e] = oub;   // out row 1
  }
}

extern "C" void kernel_launch(void* const* d_in, const int* in_sizes, int n_in,
                              void* d_out, int out_size, void* d_ws, size_t ws_size,
                              hipStream_t stream) {
  const float* bounds = (const float*)d_in[0];  // [2, N] fp32
  float* out = (float*)d_out;                   // [2, N] fp32
  const int n = in_sizes[0] / 2;                // N = 8192
  const int blocks = (n + 127) / 128;           // 128 elems per 256-thread block
  relu_bounds_wmma<<<blocks, 256, 0, stream>>>(bounds, out, n);
}